// DecodeYoloV2_56925496541420
// MI455X (gfx1250) — compile-verified
//
#include <hip/hip_runtime.h>
#include <cstdint>

#define NC      80
#define CH      85
#define HW      65536
#define KK      4096
#define DEPTH   8
#define INFLIGHT 7
#define CONF_T  0.92f
#define IOU_T   0.45f

__device__ __forceinline__ float sigmoidf_(float z) { return 1.0f / (1.0f + __expf(-z)); }

// ---- CDNA5 async global->LDS path -------------------------------------------
#define WAIT_ASYNCCNT_6 asm volatile("s_wait_asynccnt 6" ::: "memory")
#define WAIT_DSCNT_0    asm volatile("s_wait_dscnt 0"    ::: "memory")

typedef int v2i_ __attribute__((ext_vector_type(2)));
typedef __attribute__((address_space(1))) v2i_ gv2i;   // global AS pointee
typedef __attribute__((address_space(3))) v2i_ lv2i;   // LDS AS pointee

__device__ __forceinline__ void async_copy_b64(const float* gp, void* lp) {
#if defined(__has_builtin) && __has_builtin(__builtin_amdgcn_global_load_async_to_lds_b64)
  __builtin_amdgcn_global_load_async_to_lds_b64(
      (gv2i*)(unsigned long long)gp,
      (lv2i*)(unsigned)(unsigned long long)lp,
      0, 0);
#else
  asm volatile("global_load_async_to_lds_b64 %0, %1, off"
               :: "v"((unsigned)(unsigned long long)lp),
                  "v"((unsigned long long)gp)
               : "memory");
#endif
}

// ---- Kernel 1: decode (HBM-bound streaming, async-LDS 7-deep per-wave pipe) --
__global__ __launch_bounds__(256) void decode_kernel(
    const float* __restrict__ x, const float* __restrict__ anchors,
    float* __restrict__ cand_score, unsigned* __restrict__ cand_n,
    float4* __restrict__ cand_box, float* __restrict__ cand_cls,
    unsigned* __restrict__ counter,
    float* __restrict__ aux_conf, float* __restrict__ aux_x2, float* __restrict__ aux_y2)
{
  __shared__ float2 stage[8][DEPTH][32];   // [wave][slot][lane]
  __shared__ float2 dstage[8][32];         // dummy sink for constant-wait tail
  const int tid  = threadIdx.x;
  const int wv   = tid >> 5, ln = tid & 31;
  const int a    = blockIdx.x >> 7;        // 5 anchor slices
  const int tile = blockIdx.x & 127;       // 128 tiles x 512 positions
  const int pos  = tile * 512 + tid * 2;   // two spatial positions per thread
  const float* base = x + (size_t)a * CH * HW + pos;

  for (int c = 0; c < INFLIGHT; ++c)
    async_copy_b64(base + (size_t)c * HW, &stage[wv][c & (DEPTH - 1)][ln]);

  float tx[2] = {0, 0}, ty[2] = {0, 0}, tw2[2] = {0, 0}, th2[2] = {0, 0}, tc2[2] = {0, 0};
  float best[2] = {-3.402823466e38f, -3.402823466e38f};
  int   bi[2]   = {0, 0};

  for (int c = 0; c < CH; ++c) {
    WAIT_ASYNCCNT_6;                                   // oldest stage (c) landed
    float2 v = stage[wv][c & (DEPTH - 1)][ln];
    WAIT_DSCNT_0;                                      // ds read retired before slot reuse
    int cn = c + INFLIGHT;                             // issue into slot (c-1)&7
    if (cn < CH) async_copy_b64(base + (size_t)cn * HW, &stage[wv][cn & (DEPTH - 1)][ln]);
    else         async_copy_b64(base + (size_t)(CH - 1) * HW, &dstage[wv][ln]);

    if      (c == 0) { tx[0] = v.x;  tx[1] = v.y;  }
    else if (c == 1) { ty[0] = v.x;  ty[1] = v.y;  }
    else if (c == 2) { tw2[0] = v.x; tw2[1] = v.y; }
    else if (c == 3) { th2[0] = v.x; th2[1] = v.y; }
    else if (c == 4) { tc2[0] = v.x; tc2[1] = v.y; }
    else {
      if (v.x > best[0]) { best[0] = v.x; bi[0] = c - 5; }
      if (v.y > best[1]) { best[1] = v.y; bi[1] = c - 5; }
    }
  }

  const float aw = anchors[2 * a] * 8.0f, ah = anchors[2 * a + 1] * 8.0f;
  const int gx0 = pos & 255, gy0 = pos >> 8;
#pragma unroll
  for (int q = 0; q < 2; ++q) {
    float cx   = (sigmoidf_(tx[q]) + (float)(gx0 + q)) * 8.0f;
    float cy   = (sigmoidf_(ty[q]) + (float)gy0) * 8.0f;
    float bw   = __expf(tw2[q]) * aw;
    float bh   = __expf(th2[q]) * ah;
    float conf = sigmoidf_(tc2[q]);
    unsigned n = (unsigned)a * HW + (unsigned)(pos + q);
    if (conf > CONF_T) {
      unsigned slot = atomicAdd(counter, 1u);
      if (slot < KK) {
        cand_score[slot] = conf;
        cand_n[slot]     = n;
        cand_box[slot]   = make_float4(cx, cy, bw, bh);
        cand_cls[slot]   = (float)bi[q];
      }
    }
    if (n < 8192u) {  // data needed to replicate top_k padding's effect on max(x2,y2)
      aux_conf[n] = conf;
      aux_x2[n]   = cx + bw * 0.5f;
      aux_y2[n]   = cy + bh * 0.5f;
    }
  }
}

// ---- Kernel 2: sort candidates (top_k semantics), gather, offset coords ------
__global__ __launch_bounds__(1024) void prep_kernel(
    const float* __restrict__ cand_score, const unsigned* __restrict__ cand_n,
    const float4* __restrict__ cand_box, const float* __restrict__ cand_cls,
    const unsigned* __restrict__ counter,
    const float* __restrict__ aux_conf, const float* __restrict__ aux_x2,
    const float* __restrict__ aux_y2,
    float4* __restrict__ sel_box, float* __restrict__ sel_score, float* __restrict__ sel_cls,
    float* __restrict__ ox1, float* __restrict__ oy1, float* __restrict__ ox2,
    float* __restrict__ oy2, float* __restrict__ oarea)
{
  __shared__ unsigned long long key[KK];
  __shared__ unsigned val[KK];
  __shared__ unsigned scan[1024];
  __shared__ float red[1024];
  const int tid = threadIdx.x;

  unsigned count = *counter; if (count > KK) count = KK;

  for (int s = tid; s < KK; s += 1024) {
    if (s < (int)count) {
      key[s] = ((unsigned long long)__float_as_uint(cand_score[s]) << 32)
             | (unsigned)(~cand_n[s]);           // desc score, asc index tiebreak
      val[s] = (unsigned)s;
    } else { key[s] = 0ull; val[s] = 0u; }
  }
  // bitonic sort, descending
  for (int k = 2; k <= KK; k <<= 1)
    for (int j = k >> 1; j > 0; j >>= 1) {
      __syncthreads();
      for (int t = tid; t < KK; t += 1024) {
        int p = t ^ j;
        if (p > t) {
          bool desc = ((t & k) == 0);
          unsigned long long a = key[t], b = key[p];
          if ((a < b) == desc) {
            key[t] = b; key[p] = a;
            unsigned tv = val[t]; val[t] = val[p]; val[p] = tv;
          }
        }
      }
    }
  __syncthreads();

  float m = -3.402823466e38f;
  for (int s = tid; s < KK; s += 1024) {
    float4 bx; float sc, cf;
    if (s < (int)count) {
      unsigned slot = val[s];
      bx = cand_box[slot]; sc = cand_score[slot]; cf = cand_cls[slot];
      m = fmaxf(m, fmaxf(bx.x + bx.z * 0.5f, bx.y + bx.w * 0.5f));
    } else { bx = make_float4(0.f, 0.f, 0.f, 0.f); sc = 0.f; cf = 0.f; }
    sel_box[s] = bx; sel_score[s] = sc; sel_cls[s] = cf;
  }

  // fold in x2/y2 of the (KK-count) smallest below-threshold indices (top_k padding)
  unsigned need = (count < KK) ? (KK - count) : 0u;
  unsigned carry = 0;
  for (int chv = 0; chv < 8; ++chv) {
    int n = chv * 1024 + tid;
    unsigned flag = (aux_conf[n] > CONF_T) ? 0u : 1u;
    __syncthreads();
    scan[tid] = flag;
    __syncthreads();
    for (int ofs = 1; ofs < 1024; ofs <<= 1) {
      unsigned vv = (tid >= ofs) ? scan[tid - ofs] : 0u;
      __syncthreads();
      scan[tid] += vv;
      __syncthreads();
    }
    unsigned rank = carry + scan[tid] - flag;
    if (flag && rank < need) m = fmaxf(m, fmaxf(aux_x2[n], aux_y2[n]));
    carry += scan[1023];
  }

  __syncthreads();
  red[tid] = m; __syncthreads();
  for (int ofs = 512; ofs; ofs >>= 1) {
    if (tid < ofs) red[tid] = fmaxf(red[tid], red[tid + ofs]);
    __syncthreads();
  }
  const float off_unit = red[0] + 1.0f;

  for (int s = tid; s < KK; s += 1024) {
    float4 bx = sel_box[s];
    float  o  = sel_cls[s] * off_unit;
    float x1 = bx.x - bx.z * 0.5f + o, x2 = bx.x + bx.z * 0.5f + o;
    float y1 = bx.y - bx.w * 0.5f + o, y2 = bx.y + bx.w * 0.5f + o;
    ox1[s] = x1; oy1[s] = y1; ox2[s] = x2; oy2[s] = y2;
    oarea[s] = (x2 - x1) * (y2 - y1);
  }
}

// ---- Kernel 3: suppression bitmask (K x K/32), LDS-tiled ---------------------
__global__ __launch_bounds__(256) void mask_kernel(
    const float* __restrict__ ox1, const float* __restrict__ oy1,
    const float* __restrict__ ox2, const float* __restrict__ oy2,
    const float* __restrict__ oarea, unsigned* __restrict__ mask)
{
  __shared__ float jx1[256], jy1[256], jx2[256], jy2[256], ja[256];
  const int tid = threadIdx.x;
  const int i   = blockIdx.x * 256 + tid;
  const float ix1 = ox1[i], iy1 = oy1[i], ix2 = ox2[i], iy2 = oy2[i], ia = oarea[i];
  const int jt0 = blockIdx.y * 4;
  for (int jt = jt0; jt < jt0 + 4; ++jt) {
    __syncthreads();
    const int j0 = jt * 256;
    jx1[tid] = ox1[j0 + tid]; jy1[tid] = oy1[j0 + tid];
    jx2[tid] = ox2[j0 + tid]; jy2[tid] = oy2[j0 + tid];
    ja[tid]  = oarea[j0 + tid];
    __syncthreads();
    for (int w = 0; w < 8; ++w) {
      unsigned bits = 0u;
#pragma unroll 4
      for (int b = 0; b < 32; ++b) {
        int jj = w * 32 + b;
        float iw = fmaxf(fminf(ix2, jx2[jj]) - fmaxf(ix1, jx1[jj]), 0.f);
        float ih = fmaxf(fminf(iy2, jy2[jj]) - fmaxf(iy1, jy1[jj]), 0.f);
        float inter = iw * ih;
        float iou = inter / (ia + ja[jj] - inter + 1e-9f);
        if (iou > IOU_T) bits |= (1u << b);
      }
      mask[(size_t)i * 128 + (size_t)(jt * 8 + w)] = bits;
    }
  }
}

// ---- Kernel 4: greedy scan, single wave32 (lane l owns 4 of 128 LDS words) ---
__global__ __launch_bounds__(32) void nms_greedy_kernel(
    const float* __restrict__ sel_score, const unsigned* __restrict__ mask,
    unsigned* __restrict__ keepw)
{
  volatile __shared__ unsigned sup[128];
  const int ln = threadIdx.x;
  for (int r = 0; r < 4; ++r) sup[ln * 4 + r] = 0u;
  __syncthreads();
  unsigned curw = 0u;
  for (int i = 0; i < KK; ++i) {
    unsigned sw = sup[i >> 5];
    bool keep = (sel_score[i] > CONF_T) && !((sw >> (i & 31)) & 1u);
    if (keep) {
      const unsigned* mrow = mask + (size_t)i * 128;
#pragma unroll
      for (int r = 0; r < 4; ++r) sup[ln * 4 + r] |= mrow[ln * 4 + r];
    }
    curw |= (keep ? 1u : 0u) << (i & 31);
    if ((i & 31) == 31) { if (ln == 0) keepw[i >> 5] = curw; curw = 0u; }
    __syncthreads();
  }
}

// ---- Kernel 5: emit (K,6) rows, zeroing non-kept -----------------------------
__global__ __launch_bounds__(256) void write_out_kernel(
    const float4* __restrict__ sel_box, const float* __restrict__ sel_score,
    const float* __restrict__ sel_cls, const unsigned* __restrict__ keepw,
    float* __restrict__ out)
{
  const int s = blockIdx.x * 256 + threadIdx.x;
  const bool keep = (keepw[s >> 5] >> (s & 31)) & 1u;
  float4 b = sel_box[s];
  float* o = out + (size_t)s * 6;
  o[0] = keep ? b.x : 0.f; o[1] = keep ? b.y : 0.f;
  o[2] = keep ? b.z : 0.f; o[3] = keep ? b.w : 0.f;
  o[4] = keep ? sel_score[s] : 0.f;
  o[5] = keep ? sel_cls[s] : 0.f;
}

extern "C" void kernel_launch(void* const* d_in, const int* in_sizes, int n_in,
                              void* d_out, int out_size, void* d_ws, size_t ws_size,
                              hipStream_t stream)
{
  (void)in_sizes; (void)n_in; (void)out_size; (void)ws_size;
  const float* x       = (const float*)d_in[0];
  const float* anchors = (const float*)d_in[1];
  float*       out     = (float*)d_out;
  char*        ws      = (char*)d_ws;

  size_t o = 0;
  auto alloc = [&](size_t b) { size_t r = o; o += (b + 255) & ~(size_t)255; return r; };
  unsigned* counter    = (unsigned*)(ws + alloc(4));
  float*    cand_score = (float*)   (ws + alloc(4  * KK));
  unsigned* cand_n     = (unsigned*)(ws + alloc(4  * KK));
  float4*   cand_box   = (float4*)  (ws + alloc(16 * KK));
  float*    cand_cls   = (float*)   (ws + alloc(4  * KK));
  float*    aux_conf   = (float*)   (ws + alloc(4  * 8192));
  float*    aux_x2     = (float*)   (ws + alloc(4  * 8192));
  float*    aux_y2     = (float*)   (ws + alloc(4  * 8192));
  float4*   sel_box    = (float4*)  (ws + alloc(16 * KK));
  float*    sel_score  = (float*)   (ws + alloc(4  * KK));
  float*    sel_cls    = (float*)   (ws + alloc(4  * KK));
  float*    ox1        = (float*)   (ws + alloc(4  * KK));
  float*    oy1        = (float*)   (ws + alloc(4  * KK));
  float*    ox2        = (float*)   (ws + alloc(4  * KK));
  float*    oy2        = (float*)   (ws + alloc(4  * KK));
  float*    oarea      = (float*)   (ws + alloc(4  * KK));
  unsigned* mask       = (unsigned*)(ws + alloc(4  * 128 * KK));
  unsigned* keepw      = (unsigned*)(ws + alloc(4  * 128));

  (void)hipMemsetAsync(counter, 0, sizeof(unsigned), stream);
  decode_kernel<<<640, 256, 0, stream>>>(x, anchors, cand_score, cand_n, cand_box,
                                         cand_cls, counter, aux_conf, aux_x2, aux_y2);
  prep_kernel<<<1, 1024, 0, stream>>>(cand_score, cand_n, cand_box, cand_cls, counter,
                                      aux_conf, aux_x2, aux_y2,
                                      sel_box, sel_score, sel_cls,
                                      ox1, oy1, ox2, oy2, oarea);
  mask_kernel<<<dim3(16, 4), 256, 0, stream>>>(ox1, oy1, ox2, oy2, oarea, mask);
  nms_greedy_kernel<<<1, 32, 0, stream>>>(sel_score, mask, keepw);
  write_out_kernel<<<16, 256, 0, stream>>>(sel_box, sel_score, sel_cls, keepw, out);
}